// RN_30331059045069
// MI455X (gfx1250) — compile-verified
//
#include <hip/hip_runtime.h>

// Problem constants (match reference)
#define DD        256     // feature dim
#define SEQN      128     // sequence length
#define NBATCH    32
#define XE_STRIDE 260     // padded LDS stride for Xe (even -> b64-aligned pairs)
#define P_STRIDE  132     // padded LDS stride for Xl/Xr

typedef float v2f __attribute__((ext_vector_type(2)));
typedef float v8f __attribute__((ext_vector_type(8)));

// LDS float-index offsets
#define XE_OFF  0
#define XL_OFF  (SEQN * XE_STRIDE)                 // 33280
#define XR_OFF  (XL_OFF + SEQN * P_STRIDE)         // 50176
#define TOK_OFF (XR_OFF + SEQN * P_STRIDE)         // 67072
#define SMEM_FLOATS (TOK_OFF + SEQN)               // 67200 -> 268800 bytes

// ---------------------------------------------------------------------------
// Kernel A: per (batch, d-half) workgroup.
//   1) gather Xe = emb[X[b]]          -> LDS
//   2) Xl = Xe@W_l^T+b_l, Xr likewise -> LDS   (V_WMMA_F32_16X16X4_F32)
//      register-blocked: one wave owns a 16-wide d column unit, keeps 8
//      row-tile accumulators live, and reuses each B fragment 8x.
//   3) pooled_part[d] = sum_{i,j} relu(Xl[j,d] + Xr[i,d])  (all in LDS)
// ---------------------------------------------------------------------------
__global__ __launch_bounds__(256) void RN_pair_fused_kernel(
    const int* __restrict__ X, const float* __restrict__ emb,
    const float* __restrict__ W_l, const float* __restrict__ b_l,
    const float* __restrict__ W_r, const float* __restrict__ b_r,
    float* __restrict__ pooled) {
  extern __shared__ float smem[];
  float* xe = smem + XE_OFF;              // [128][260]
  float* xl = smem + XL_OFF;              // [128][132]
  float* xr = smem + XR_OFF;              // [128][132]
  int*   tok = (int*)(smem + TOK_OFF);    // [128]

  const int b     = blockIdx.x;           // batch
  const int dbase = blockIdx.y * 128;     // which 128 feature columns
  const int tid   = threadIdx.x;

  // --- stage tokens ---
  if (tid < SEQN) tok[tid] = X[b * SEQN + tid];
  __syncthreads();

  // --- gather embedding rows into LDS (float4 granularity) ---
  for (int idx = tid; idx < SEQN * (DD / 4); idx += 256) {
    const int row = idx >> 6;             // 64 float4 per row
    const int c4  = idx & 63;
    const float4 v = ((const float4*)(emb + (long)tok[row] * DD))[c4];
    float* dst = xe + row * XE_STRIDE + c4 * 4;
    dst[0] = v.x; dst[1] = v.y; dst[2] = v.z; dst[3] = v.w;
  }
  __syncthreads();

  // --- WMMA GEMMs: 16 column-units (8 d-tiles x {Xl,Xr}) over 8 waves ---
  const int wave = tid >> 5;
  const int lane = tid & 31;
  const int lr   = lane & 15;             // row (A) / col (B) within tile
  const int koff = (lane >> 4) * 2;       // k sub-offset per 16-lane half

  for (int cu = wave; cu < 16; cu += 8) {
    const int mat     = cu >> 3;          // 0 -> Xl, 1 -> Xr
    const int colbase = (cu & 7) * 16;    // d tile (within this 128-half)
    const float* W    = mat ? W_r : W_l;
    const float* bias = mat ? b_r : b_l;
    float*       dst  = mat ? xr : xl;
    const int dcol    = dbase + colbase + lr;

    const float bv = bias[dcol];
    v8f acc[8];
#pragma unroll
    for (int rt = 0; rt < 8; ++rt)
      acc[rt] = (v8f){ bv, bv, bv, bv, bv, bv, bv, bv };

    const float* brow = W + (long)dcol * DD + koff;   // global, 8B aligned
    for (int kb = 0; kb < DD; kb += 32) {             // K-blocks of 32
      // B fragments for this K-block: loaded once, reused by 8 row tiles
      v2f bf[8];
#pragma unroll
      for (int s = 0; s < 8; ++s)
        bf[s] = *(const v2f*)(brow + kb + 4 * s);
#pragma unroll
      for (int rt = 0; rt < 8; ++rt) {
        const float* arow = xe + (rt * 16 + lr) * XE_STRIDE + kb + koff;
#pragma unroll
        for (int s = 0; s < 8; ++s) {
          v2f a = *(const v2f*)(arow + 4 * s);        // LDS, 8B aligned
          acc[rt] = __builtin_amdgcn_wmma_f32_16x16x4_f32(
              /*neg_a=*/false, a, /*neg_b=*/false, bf[s],
              /*c_mod=*/(short)0, acc[rt],
              /*reuse_a=*/false, /*reuse_b=*/false);
        }
      }
    }
    // C/D layout: VGPR v, lanes 0-15 -> M=v, lanes 16-31 -> M=v+8, N=lane%16
#pragma unroll
    for (int rt = 0; rt < 8; ++rt) {
      float* obase = dst + (rt * 16 + 8 * (lane >> 4)) * P_STRIDE + colbase + lr;
#pragma unroll
      for (int v = 0; v < 8; ++v) obase[v * P_STRIDE] = acc[rt][v];
    }
  }
  __syncthreads();

  // --- pair reduction: thread owns (d, half of i-range), all j ---
  const int d    = tid & 127;
  const int ih0  = (tid >> 7) * 64;       // i half: [ih0, ih0+64)
  float acc = 0.f;
  for (int jb = 0; jb < SEQN; jb += 8) {
    float xlv[8];
#pragma unroll
    for (int u = 0; u < 8; ++u) xlv[u] = xl[(jb + u) * P_STRIDE + d];
    for (int i = ih0; i < ih0 + 64; ++i) {
      const float xrv = xr[i * P_STRIDE + d];
#pragma unroll
      for (int u = 0; u < 8; ++u) acc += fmaxf(xlv[u] + xrv, 0.f);
    }
  }
  // combine the two i-halves via LDS (reuse xe region; xl/xr untouched)
  smem[tid] = acc;
  __syncthreads();
  if (tid < 128)
    pooled[b * DD + dbase + tid] = smem[tid] + smem[tid + 128];
}

// ---------------------------------------------------------------------------
// Kernel B: out = pooled @ W_rn^T + 128*128*b_rn.  One wave per 16x16 tile.
// ---------------------------------------------------------------------------
__global__ __launch_bounds__(32) void RN_out_kernel(
    const float* __restrict__ pooled, const float* __restrict__ W_rn,
    const float* __restrict__ b_rn, float* __restrict__ out) {
  const int mt   = blockIdx.x;            // 0..1  (batch tiles, M=32)
  const int nt   = blockIdx.y;            // 0..15 (feature tiles, N=256)
  const int lane = threadIdx.x;
  const int lr   = lane & 15;
  const int koff = (lane >> 4) * 2;
  const int row  = mt * 16 + lr;          // batch row for A
  const int col  = nt * 16 + lr;          // output feature for B

  const float bv = 16384.0f * b_rn[col];  // n*n*b_rn, n=128
  v8f c = { bv, bv, bv, bv, bv, bv, bv, bv };

  const float* arow = pooled + (long)row * DD + koff;
  const float* brow = W_rn + (long)col * DD + koff;
  for (int k0 = 0; k0 < DD; k0 += 4) {
    v2f a  = *(const v2f*)(arow + k0);
    v2f bb = *(const v2f*)(brow + k0);
    c = __builtin_amdgcn_wmma_f32_16x16x4_f32(
        false, a, false, bb, (short)0, c, false, false);
  }
  float* obase = out + (mt * 16 + 8 * (lane >> 4)) * DD + nt * 16 + lr;
#pragma unroll
  for (int v = 0; v < 8; ++v) obase[v * DD] = c[v];
}

// ---------------------------------------------------------------------------
extern "C" void kernel_launch(void* const* d_in, const int* in_sizes, int n_in,
                              void* d_out, int out_size, void* d_ws, size_t ws_size,
                              hipStream_t stream) {
  (void)in_sizes; (void)n_in; (void)out_size; (void)ws_size;
  const int*   X    = (const int*)d_in[0];
  const float* emb  = (const float*)d_in[1];
  const float* W_l  = (const float*)d_in[2];
  const float* b_l  = (const float*)d_in[3];
  const float* W_r  = (const float*)d_in[4];
  const float* b_r  = (const float*)d_in[5];
  const float* W_rn = (const float*)d_in[6];
  const float* b_rn = (const float*)d_in[7];
  float* pooled = (float*)d_ws;           // 32*256 f32 scratch
  float* out    = (float*)d_out;

  dim3 gridA(NBATCH, 2), blockA(256);
  size_t shmem = (size_t)SMEM_FLOATS * sizeof(float);   // 268,800 B <= 320 KB/WGP
  RN_pair_fused_kernel<<<gridA, blockA, shmem, stream>>>(
      X, emb, W_l, b_l, W_r, b_r, pooled);

  dim3 gridB(2, 16), blockB(32);
  RN_out_kernel<<<gridB, blockB, 0, stream>>>(pooled, W_rn, b_rn, out);
}